// Net_69887707840818
// MI455X (gfx1250) — compile-verified
//
#include <hip/hip_runtime.h>
#include <math.h>

// ---------------------------------------------------------------------------
// MI455X (gfx1250, wave32). All GEMMs via V_WMMA_F32_16X16X4_F32 (fp32 path:
// workload is only ~180 GFLOP and intermediates are L2-resident, so fp32 WMMA
// is already far from any bandwidth/compute wall; keeps reference accuracy).
// Pipeline: prep(transpose weights) -> inproj -> tcn x3 -> post -> attn ->
//           gru -> head.
// d_ws: activations (~303 MB) + transposed weights (~0.9 MB).
// ---------------------------------------------------------------------------

typedef float v2f __attribute__((ext_vector_type(2)));
typedef float v4f __attribute__((ext_vector_type(4)));
typedef float v8f __attribute__((ext_vector_type(8)));

#define BATCH  4096
#define SEQL   72
#define DM     96
#define CT     64
#define HIDG   48

// D = A(16x4) * B(4x16) + C, fp32. Layouts (ISA 7.12.2):
//   A: lane l(0..15) row=l, K=k0..k0+1; lanes 16..31 K=k0+2..k0+3
//   B: symmetric, lane = column
//   C: reg r -> row = r + 8*(lane>=16), col = lane&15
__device__ __forceinline__ v8f wmma4(v2f a, v2f b, v8f c) {
  return __builtin_amdgcn_wmma_f32_16x16x4_f32(false, a, false, b, (short)0, c,
                                               false, false);
}
__device__ __forceinline__ v2f ld2(const float* p) { return *(const v2f*)p; }
__device__ __forceinline__ v4f ld4(const float* p) { return *(const v4f*)p; }

__device__ __forceinline__ float geluf(float x) {
  return 0.5f * x * (1.0f + erff(x * 0.7071067811865476f));
}
__device__ __forceinline__ float sigmf(float x) {
  return 1.0f / (1.0f + expf(-x));
}

// ---------------------------------------------------------------------------
// Weight prep: transposes so B-fragments are contiguous (global_load_b64).
// ---------------------------------------------------------------------------
__global__ __launch_bounds__(256) void k_tr2d(const float* __restrict__ src,
                                              float* __restrict__ dst, int K, int N) {
  int i = blockIdx.x * blockDim.x + threadIdx.x;
  if (i >= K * N) return;
  int n = i % N, k = i / N;
  dst[n * K + k] = src[i];                 // dst[N][K] = src[K][N]^T
}
// conv weight [O][I][5] -> [O][5][I]
__global__ __launch_bounds__(256) void k_trconv(const float* __restrict__ src,
                                                float* __restrict__ dst, int O, int I) {
  int idx = blockIdx.x * blockDim.x + threadIdx.x;
  if (idx >= O * I * 5) return;
  int t = idx % 5, rem = idx / 5, i = rem % I, o = rem / I;
  dst[(o * 5 + t) * I + i] = src[idx];
}

// ---------------------------------------------------------------------------
// Kernel 1: input projection [BL,6] @ [6,96] + bias (K=6 -> VALU)
// ---------------------------------------------------------------------------
__global__ __launch_bounds__(256) void k_inproj(
    const float* __restrict__ x, const float* __restrict__ inw,
    const float* __restrict__ inb, float* __restrict__ z, int ntok) {
  int t = blockIdx.x * blockDim.x + threadIdx.x;
  if (t >= ntok) return;
  float xi[6];
#pragma unroll
  for (int i = 0; i < 6; i++) xi[i] = x[t * 6 + i];
  float* zr = z + (size_t)t * DM;
  for (int o = 0; o < DM; o += 4) {
    v4f acc = ld4(inb + o);
#pragma unroll
    for (int i = 0; i < 6; i++) acc += xi[i] * ld4(inw + i * DM + o);
    *(v4f*)(zr + o) = acc;
  }
}

// ---------------------------------------------------------------------------
// Kernel 2: TCN block, one sequence per WG (4 waves).  Causal dilated conv =
// 5 tap-shifted WMMA GEMMs.  LDS has a 16-row zero halo so every A load is an
// unconditional ds_load_b64 (no EXEC toggling).  Weights pre-transposed to
// [O][5][I] so B loads are contiguous b64.
// ---------------------------------------------------------------------------
template <int CIN, int D1, int D2, bool HASRES>
__global__ __launch_bounds__(128) void k_tcn(
    const float* __restrict__ in, float* __restrict__ out,
    const float* __restrict__ ng, const float* __restrict__ nb,
    const float* __restrict__ w1t, const float* __restrict__ b1,
    const float* __restrict__ w2t, const float* __restrict__ b2,
    const float* __restrict__ wres, const float* __restrict__ bres) {
  constexpr int PAD = 16;
  extern __shared__ float smem[];
  float* hln = smem;                          // [PAD+72][CIN], rows<PAD zero
  float* y1  = smem + (PAD + SEQL) * CIN;     // [PAD+80][64],  rows<PAD zero
  const int seq = blockIdx.x;
  const float* inp = in + (size_t)seq * SEQL * CIN;

  for (int i = threadIdx.x; i < PAD * CIN; i += blockDim.x) hln[i] = 0.f;
  for (int i = threadIdx.x; i < PAD * CT; i += blockDim.x) y1[i] = 0.f;

  // LayerNorm rows (vectorized b128)
  for (int t = threadIdx.x; t < SEQL; t += blockDim.x) {
    const float* r = inp + t * CIN;
    v4f s4 = {0.f, 0.f, 0.f, 0.f};
#pragma unroll
    for (int i = 0; i < CIN; i += 4) s4 += ld4(r + i);
    float m = (s4.x + s4.y + s4.z + s4.w) / (float)CIN;
    v4f q4 = {0.f, 0.f, 0.f, 0.f};
#pragma unroll
    for (int i = 0; i < CIN; i += 4) { v4f d = ld4(r + i) - m; q4 += d * d; }
    float inv = rsqrtf((q4.x + q4.y + q4.z + q4.w) / (float)CIN + 1e-5f);
    float* hp = hln + (PAD + t) * CIN;
#pragma unroll
    for (int i = 0; i < CIN; i += 4)
      *(v4f*)(hp + i) = (ld4(r + i) - m) * inv * ld4(ng + i) + ld4(nb + i);
  }
  __syncthreads();

  const int lane = threadIdx.x & 31, wid = threadIdx.x >> 5, nwv = blockDim.x >> 5;
  const int col = lane & 15, koff = (lane >> 4) << 1;
  const int row16 = lane & 15, crow0 = (lane >> 4) << 3;

  // conv1: [80,CIN]x[CIN,64], jobs = 5 Mtiles x 2 Npairs
  for (int job = wid; job < 10; job += nwv) {
    int mt = job >> 1, np = job & 1;
    int co0 = np * 32 + col, co1 = co0 + 16;
    v8f a0, a1;
#pragma unroll
    for (int r = 0; r < 8; r++) { a0[r] = b1[co0]; a1[r] = b1[co1]; }
    int row = mt * 16 + row16;
    int rowc = row < SEQL ? row : SEQL - 1;
#pragma unroll
    for (int j = 0; j < 5; j++) {
      int kw = 4 - j;
      const float* arow = hln + (rowc - j * D1 + PAD) * CIN;
      const float* wp0 = w1t + (co0 * 5 + kw) * CIN;
      const float* wp1 = w1t + (co1 * 5 + kw) * CIN;
#pragma unroll 4
      for (int k0 = 0; k0 < CIN; k0 += 4) {
        v2f a = ld2(arow + k0 + koff);
        a0 = wmma4(a, ld2(wp0 + k0 + koff), a0);
        a1 = wmma4(a, ld2(wp1 + k0 + koff), a1);
      }
    }
#pragma unroll
    for (int r = 0; r < 8; r++) {       // unconditional store into padded y1
      int m = mt * 16 + crow0 + r;
      y1[(PAD + m) * CT + co0] = geluf(a0[r]);
      y1[(PAD + m) * CT + co1] = geluf(a1[r]);
    }
  }
  __syncthreads();

  // conv2 + residual -> global
  float* outp = out + (size_t)seq * SEQL * CT;
  for (int job = wid; job < 10; job += nwv) {
    int mt = job >> 1, np = job & 1;
    int co0 = np * 32 + col, co1 = co0 + 16;
    v8f a0, a1;
#pragma unroll
    for (int r = 0; r < 8; r++) { a0[r] = b2[co0]; a1[r] = b2[co1]; }
    int row = mt * 16 + row16;
    int rowc = row < SEQL ? row : SEQL - 1;
#pragma unroll
    for (int j = 0; j < 5; j++) {
      int kw = 4 - j;
      const float* arow = y1 + (rowc - j * D2 + PAD) * CT;
      const float* wp0 = w2t + (co0 * 5 + kw) * CT;
      const float* wp1 = w2t + (co1 * 5 + kw) * CT;
#pragma unroll 4
      for (int k0 = 0; k0 < CT; k0 += 4) {
        v2f a = ld2(arow + k0 + koff);
        a0 = wmma4(a, ld2(wp0 + k0 + koff), a0);
        a1 = wmma4(a, ld2(wp1 + k0 + koff), a1);
      }
    }
    v8f r0, r1;
    if constexpr (HASRES) {             // 1x1 conv residual, wres already [O][I]
#pragma unroll
      for (int r = 0; r < 8; r++) { r0[r] = bres[co0]; r1[r] = bres[co1]; }
      const float* arow = hln + (rowc + PAD) * CIN;
#pragma unroll 4
      for (int k0 = 0; k0 < CIN; k0 += 4) {
        v2f a = ld2(arow + k0 + koff);
        r0 = wmma4(a, ld2(wres + co0 * CIN + k0 + koff), r0);
        r1 = wmma4(a, ld2(wres + co1 * CIN + k0 + koff), r1);
      }
    }
    if (mt < 4) {                       // wave-uniform: all 16 rows valid
#pragma unroll
      for (int r = 0; r < 8; r++) {
        int m = mt * 16 + crow0 + r;
        float b0v, b1v;
        if constexpr (HASRES) { b0v = r0[r]; b1v = r1[r]; }
        else { b0v = hln[(PAD + m) * CIN + co0]; b1v = hln[(PAD + m) * CIN + co1]; }
        outp[m * CT + co0] = geluf(a0[r]) + b0v;
        outp[m * CT + co1] = geluf(a1[r]) + b1v;
      }
    } else {
#pragma unroll
      for (int r = 0; r < 8; r++) {
        int m = mt * 16 + crow0 + r;
        if (m < SEQL) {
          float b0v, b1v;
          if constexpr (HASRES) { b0v = r0[r]; b1v = r1[r]; }
          else { b0v = hln[(PAD + m) * CIN + co0]; b1v = hln[(PAD + m) * CIN + co1]; }
          outp[m * CT + co0] = geluf(a0[r]) + b0v;
          outp[m * CT + co1] = geluf(a1[r]) + b1v;
        }
      }
    }
  }
}

// ---------------------------------------------------------------------------
// Kernel 3: post projection [BL,64] @ [64,96], one M-tile per wave, K-outer
// with 6 N-accumulators. postT is [96][64].
// ---------------------------------------------------------------------------
__global__ __launch_bounds__(256) void k_post(const float* __restrict__ hin,
                                              const float* __restrict__ wT,
                                              const float* __restrict__ bias,
                                              float* __restrict__ out) {
  extern __shared__ float sAll[];
  const int lane = threadIdx.x & 31, wid = threadIdx.x >> 5;
  const int mt = blockIdx.x * 8 + wid;
  float* As = sAll + wid * (16 * CT);
  for (int i = lane * 4; i < 16 * CT; i += 128)
    *(v4f*)(As + i) = ld4(hin + (size_t)mt * (16 * CT) + i);
  const int col = lane & 15, koff = (lane >> 4) << 1;
  const int row16 = lane & 15, crow0 = (lane >> 4) << 3;
  v8f acc[6];
#pragma unroll
  for (int nt = 0; nt < 6; nt++)
#pragma unroll
    for (int r = 0; r < 8; r++) acc[nt][r] = bias[nt * 16 + col];
#pragma unroll 4
  for (int k0 = 0; k0 < CT; k0 += 4) {
    v2f a = ld2(As + row16 * CT + k0 + koff);
#pragma unroll
    for (int nt = 0; nt < 6; nt++)
      acc[nt] = wmma4(a, ld2(wT + (nt * 16 + col) * CT + k0 + koff), acc[nt]);
  }
#pragma unroll
  for (int nt = 0; nt < 6; nt++)
#pragma unroll
    for (int r = 0; r < 8; r++)
      out[(size_t)(mt * 16 + crow0 + r) * DM + nt * 16 + col] = acc[nt][r];
}

// ---------------------------------------------------------------------------
// LayerNorm of 16 rows x 96 cols using all 32 lanes (2 lanes/row).
// ---------------------------------------------------------------------------
__device__ __forceinline__ void ln_rows16(const float* __restrict__ src,
                                          float* __restrict__ dst,
                                          const float* __restrict__ g,
                                          const float* __restrict__ bt, int lane) {
  const int row = lane & 15, half = lane >> 4;
  const float* r = src + row * DM + half * 48;
  v4f s4 = {0.f, 0.f, 0.f, 0.f};
#pragma unroll
  for (int i = 0; i < 48; i += 4) s4 += ld4(r + i);
  float s = s4.x + s4.y + s4.z + s4.w;
  s += __shfl_xor(s, 16);
  const float m = s * (1.f / DM);
  v4f q4 = {0.f, 0.f, 0.f, 0.f};
#pragma unroll
  for (int i = 0; i < 48; i += 4) { v4f d = ld4(r + i) - m; q4 += d * d; }
  float v = q4.x + q4.y + q4.z + q4.w;
  v += __shfl_xor(v, 16);
  const float inv = rsqrtf(v * (1.f / DM) + 1e-5f);
  float* dp = dst + row * DM + half * 48;
#pragma unroll
  for (int i = 0; i < 48; i += 4)
    *(v4f*)(dp + i) = (ld4(r + i) - m) * inv * ld4(g + half * 48 + i) + ld4(bt + half * 48 + i);
}

// ---------------------------------------------------------------------------
// Kernel 4: local windowed attention. One sequence per 1-wave WG; 8 windows
// (16 rows = one WMMA M-tile each) processed serially, overlap-added in LDS.
// All weight matrices pre-transposed -> contiguous b64 B loads.
// ---------------------------------------------------------------------------
__global__ __launch_bounds__(32) void k_attn(
    const float* __restrict__ p, float* __restrict__ yout,
    const float* __restrict__ ln1g, const float* __restrict__ ln1b,
    const float* __restrict__ qkvT, const float* __restrict__ qkvb,
    const float* __restrict__ owT, const float* __restrict__ ob,
    const float* __restrict__ ln2g, const float* __restrict__ ln2b,
    const float* __restrict__ f1T, const float* __restrict__ f1b,
    const float* __restrict__ f2T, const float* __restrict__ f2b,
    const float* __restrict__ plng, const float* __restrict__ plnb) {
  __shared__ float sAcc[SEQL * DM];
  __shared__ float sS[16 * DM];     // seg -> h1
  __shared__ float sH[16 * DM];     // LN outputs
  __shared__ float sO[16 * DM];     // attn concat -> h1+f
  __shared__ float sF[16 * 192];    // FFN hidden
  __shared__ float sQ[256], sK[256], sVT[256];

  const int b = blockIdx.x;
  const int lane = threadIdx.x;
  const int col = lane & 15, koff = (lane >> 4) << 1;
  const int row16 = lane & 15, crow0 = (lane >> 4) << 3;
  const float* pb = p + (size_t)b * SEQL * DM;

  for (int i = lane * 4; i < SEQL * DM; i += 128) *(v4f*)(sAcc + i) = v4f{0.f, 0.f, 0.f, 0.f};

  for (int w = 0; w < 8; w++) {
    const int t0 = w * 8;
    for (int i = lane * 4; i < 16 * DM; i += 128)
      *(v4f*)(sS + i) = ld4(pb + t0 * DM + i);
    ln_rows16(sS, sH, ln1g, ln1b, lane);

    // ---- per-head attention ----
    for (int hh = 0; hh < 6; hh++) {
      const int c = hh * 16 + col;
      v8f aq, ak, av;
#pragma unroll
      for (int r = 0; r < 8; r++) { aq[r] = qkvb[c]; ak[r] = qkvb[96 + c]; av[r] = qkvb[192 + c]; }
      const float* wq = qkvT + (size_t)c * DM;
      const float* wk = qkvT + (size_t)(96 + c) * DM;
      const float* wv = qkvT + (size_t)(192 + c) * DM;
#pragma unroll 4
      for (int k0 = 0; k0 < DM; k0 += 4) {
        v2f a = ld2(sH + row16 * DM + k0 + koff);
        aq = wmma4(a, ld2(wq + k0 + koff), aq);
        ak = wmma4(a, ld2(wk + k0 + koff), ak);
        av = wmma4(a, ld2(wv + k0 + koff), av);
      }
#pragma unroll
      for (int r = 0; r < 8; r++) {
        sQ[(crow0 + r) * 16 + col] = aq[r];   // [token][d]
        sK[(crow0 + r) * 16 + col] = ak[r];   // [token][d]
        sVT[col * 16 + crow0 + r] = av[r];    // v^T: [d][token]
      }
      // scores = q @ k^T * 0.25
      v8f sc;
#pragma unroll
      for (int r = 0; r < 8; r++) sc[r] = 0.f;
#pragma unroll
      for (int k0 = 0; k0 < 16; k0 += 4) {
        v2f a = ld2(sQ + row16 * 16 + k0 + koff);
        sc = wmma4(a, ld2(sK + col * 16 + k0 + koff), sc);
      }
      // causal softmax (rows striped over lanes in C layout)
#pragma unroll
      for (int r = 0; r < 8; r++) {
        int m = crow0 + r;
        bool valid = (col <= m);
        float val = sc[r] * 0.25f;
        float mv = valid ? val : -3.0e38f;
        for (int off = 1; off < 16; off <<= 1) mv = fmaxf(mv, __shfl_xor(mv, off));
        float e = valid ? expf(val - mv) : 0.f;
        float ssum = e;
        for (int off = 1; off < 16; off <<= 1) ssum += __shfl_xor(ssum, off);
        sc[r] = e / ssum;
      }
#pragma unroll
      for (int r = 0; r < 8; r++) sQ[(crow0 + r) * 16 + col] = sc[r];  // P [q][key]
      v8f ovv;
#pragma unroll
      for (int r = 0; r < 8; r++) ovv[r] = 0.f;
#pragma unroll
      for (int k0 = 0; k0 < 16; k0 += 4) {
        v2f a = ld2(sQ + row16 * 16 + k0 + koff);
        ovv = wmma4(a, ld2(sVT + col * 16 + k0 + koff), ovv);
      }
#pragma unroll
      for (int r = 0; r < 8; r++) sO[(crow0 + r) * DM + hh * 16 + col] = ovv[r];
    }

    // ---- out projection + residual: h1 = seg + o (into sS) ----
    {
      v8f acc[6];
#pragma unroll
      for (int nt = 0; nt < 6; nt++)
#pragma unroll
        for (int r = 0; r < 8; r++) acc[nt][r] = ob[nt * 16 + col];
#pragma unroll 4
      for (int k0 = 0; k0 < DM; k0 += 4) {
        v2f a = ld2(sO + row16 * DM + k0 + koff);
#pragma unroll
        for (int nt = 0; nt < 6; nt++)
          acc[nt] = wmma4(a, ld2(owT + (size_t)(nt * 16 + col) * DM + k0 + koff), acc[nt]);
      }
#pragma unroll
      for (int nt = 0; nt < 6; nt++)
#pragma unroll
        for (int r = 0; r < 8; r++) sS[(crow0 + r) * DM + nt * 16 + col] += acc[nt][r];
    }
    ln_rows16(sS, sH, ln2g, ln2b, lane);

    // ---- FFN: f1 = gelu(h @ f1w + b)  [16,192] ----
    {
      v8f acc[12];
#pragma unroll
      for (int nt = 0; nt < 12; nt++)
#pragma unroll
        for (int r = 0; r < 8; r++) acc[nt][r] = f1b[nt * 16 + col];
#pragma unroll 2
      for (int k0 = 0; k0 < DM; k0 += 4) {
        v2f a = ld2(sH + row16 * DM + k0 + koff);
#pragma unroll
        for (int nt = 0; nt < 12; nt++)
          acc[nt] = wmma4(a, ld2(f1T + (size_t)(nt * 16 + col) * DM + k0 + koff), acc[nt]);
      }
#pragma unroll
      for (int nt = 0; nt < 12; nt++)
#pragma unroll
        for (int r = 0; r < 8; r++) sF[(crow0 + r) * 192 + nt * 16 + col] = geluf(acc[nt][r]);
    }
    // ---- f2 + h1 -> sO ----
    {
      v8f acc[6];
#pragma unroll
      for (int nt = 0; nt < 6; nt++)
#pragma unroll
        for (int r = 0; r < 8; r++) acc[nt][r] = f2b[nt * 16 + col];
#pragma unroll 4
      for (int k0 = 0; k0 < 192; k0 += 4) {
        v2f a = ld2(sF + row16 * 192 + k0 + koff);
#pragma unroll
        for (int nt = 0; nt < 6; nt++)
          acc[nt] = wmma4(a, ld2(f2T + (size_t)(nt * 16 + col) * 192 + k0 + koff), acc[nt]);
      }
#pragma unroll
      for (int nt = 0; nt < 6; nt++)
#pragma unroll
        for (int r = 0; r < 8; r++) {
          int m = crow0 + r;
          sO[m * DM + nt * 16 + col] = sS[m * DM + nt * 16 + col] + acc[nt][r];
        }
    }
    // ---- post-LN + overlap-add ----
    {
      const int row = lane & 15, half = lane >> 4;
      const float* r = sO + row * DM + half * 48;
      v4f s4 = {0.f, 0.f, 0.f, 0.f};
#pragma unroll
      for (int i = 0; i < 48; i += 4) s4 += ld4(r + i);
      float s = s4.x + s4.y + s4.z + s4.w;
      s += __shfl_xor(s, 16);
      const float m = s * (1.f / DM);
      v4f q4 = {0.f, 0.f, 0.f, 0.f};
#pragma unroll
      for (int i = 0; i < 48; i += 4) { v4f d = ld4(r + i) - m; q4 += d * d; }
      float v = q4.x + q4.y + q4.z + q4.w;
      v += __shfl_xor(v, 16);
      const float inv = rsqrtf(v * (1.f / DM) + 1e-5f);
      float* ap = sAcc + (t0 + row) * DM + half * 48;
#pragma unroll
      for (int i = 0; i < 48; i += 4)
        *(v4f*)(ap + i) = ld4(ap + i) + (ld4(r + i) - m) * inv * ld4(plng + half * 48 + i) +
                          ld4(plnb + half * 48 + i);
    }
  }
  // count-normalize + store (tokens [8,64) covered by two windows)
  for (int i = lane * 4; i < SEQL * DM; i += 128) {
    int t = i / DM;
    float scale = (t < 8 || t >= 64) ? 1.f : 0.5f;
    *(v4f*)(yout + (size_t)b * SEQL * DM + i) = ld4(sAcc + i) * scale;
  }
}

// ---------------------------------------------------------------------------
// Kernel 5: GRU, one wave per 16 batch rows, 72 sequential steps.  All 9 gate
// column-tiles accumulated per K pass (1 A load feeds 9 WMMAs).  wih/whh are
// [out][in] so B loads are already contiguous b64.
// ---------------------------------------------------------------------------
__global__ __launch_bounds__(32) void k_gru(
    const float* __restrict__ y, const float* __restrict__ wih,
    const float* __restrict__ whh, const float* __restrict__ bih,
    const float* __restrict__ bhh, float* __restrict__ hT) {
  __shared__ float sWhh[144 * HIDG];
  __shared__ float sY[16 * DM];
  __shared__ float sHst[16 * HIDG];
  const int lane = threadIdx.x;
  const int b0 = blockIdx.x * 16;
  const int col = lane & 15, koff = (lane >> 4) << 1;
  const int row16 = lane & 15, crow0 = (lane >> 4) << 3;

  for (int i = lane * 4; i < 144 * HIDG; i += 128) *(v4f*)(sWhh + i) = ld4(whh + i);
  for (int i = lane * 4; i < 16 * HIDG; i += 128) *(v4f*)(sHst + i) = v4f{0.f, 0.f, 0.f, 0.f};
  v8f hC[3];
#pragma unroll
  for (int nt = 0; nt < 3; nt++)
#pragma unroll
    for (int r = 0; r < 8; r++) hC[nt][r] = 0.f;

  for (int t = 0; t < SEQL; t++) {
    for (int i = lane * 4; i < 16 * DM; i += 128)
      *(v4f*)(sY + i) = ld4(y + ((size_t)(b0 + i / DM) * SEQL + t) * DM + (i % DM));

    v8f xg[9], hg[9];
#pragma unroll
    for (int g = 0; g < 9; g++)
#pragma unroll
      for (int r = 0; r < 8; r++) { xg[g][r] = bih[g * 16 + col]; hg[g][r] = bhh[g * 16 + col]; }
    for (int k0 = 0; k0 < DM; k0 += 4) {
      v2f a = ld2(sY + row16 * DM + k0 + koff);
#pragma unroll
      for (int g = 0; g < 9; g++)
        xg[g] = wmma4(a, ld2(wih + (size_t)(g * 16 + col) * DM + k0 + koff), xg[g]);
    }
    for (int k0 = 0; k0 < HIDG; k0 += 4) {
      v2f a = ld2(sHst + row16 * HIDG + k0 + koff);
#pragma unroll
      for (int g = 0; g < 9; g++)
        hg[g] = wmma4(a, ld2(sWhh + (g * 16 + col) * HIDG + k0 + koff), hg[g]);
    }
#pragma unroll
    for (int nt = 0; nt < 3; nt++) {
#pragma unroll
      for (int r = 0; r < 8; r++) {
        float rr = sigmf(xg[nt][r] + hg[nt][r]);
        float zz = sigmf(xg[3 + nt][r] + hg[3 + nt][r]);
        float nn = tanhf(xg[6 + nt][r] + rr * hg[6 + nt][r]);
        hC[nt][r] = (1.f - zz) * nn + zz * hC[nt][r];
      }
    }
#pragma unroll
    for (int nt = 0; nt < 3; nt++)
#pragma unroll
      for (int r = 0; r < 8; r++)
        sHst[(crow0 + r) * HIDG + nt * 16 + col] = hC[nt][r];
  }
  for (int i = lane * 4; i < 16 * HIDG; i += 128)
    *(v4f*)(hT + (size_t)b0 * HIDG + i) = ld4(sHst + i);
}

// ---------------------------------------------------------------------------
// Kernel 6: head  pred = gelu(h @ h1w + h1b) @ h2w + h2b
// ---------------------------------------------------------------------------
__global__ __launch_bounds__(128) void k_head(
    const float* __restrict__ hT, const float* __restrict__ h1w,
    const float* __restrict__ h1b, const float* __restrict__ h2w,
    const float* __restrict__ h2b, float* __restrict__ out) {
  int b = blockIdx.x * blockDim.x + threadIdx.x;
  if (b >= BATCH) return;
  const float* h = hT + b * HIDG;
  float acc = h2b[0];
  for (int o = 0; o < 32; o++) {
    float s = h1b[o];
#pragma unroll 8
    for (int i = 0; i < HIDG; i++) s += h[i] * h1w[i * 32 + o];
    acc += geluf(s) * h2w[o];
  }
  out[b] = acc;
}

// ---------------------------------------------------------------------------
// Input flattening (setup_inputs dict order, depth-first):
// ---------------------------------------------------------------------------
enum {
  IN_X = 0, IN_INW, IN_INB,
  T1_NG, T1_NB, T1_W1, T1_B1, T1_W2, T1_B2, T1_WRES, T1_BRES,
  T2_NG, T2_NB, T2_W1, T2_B1, T2_W2, T2_B2,
  T3_NG, T3_NB, T3_W1, T3_B1, T3_W2, T3_B2,
  IN_POSTW, IN_POSTB,
  LA_LN1G, LA_LN1B, LA_QKVW, LA_QKVB, LA_OW, LA_OB, LA_LN2G, LA_LN2B,
  LA_F1W, LA_F1B, LA_F2W, LA_F2B, LA_PLNG, LA_PLNB,
  GRU_WIH, GRU_WHH, GRU_BIH, GRU_BHH,
  H1W, H1B, H2W, H2B
};

extern "C" void kernel_launch(void* const* d_in, const int* in_sizes, int n_in,
                              void* d_out, int out_size, void* d_ws, size_t ws_size,
                              hipStream_t stream) {
  (void)in_sizes; (void)n_in; (void)out_size; (void)ws_size;
  auto F = [&](int i) { return (const float*)d_in[i]; };
  const size_t BL = (size_t)BATCH * SEQL;

  float* A    = (float*)d_ws;              // [BL*96] z, later attn out
  float* Bb   = A + BL * DM;               // [BL*64]
  float* C    = Bb + BL * CT;              // [BL*96]
  float* hT   = C + BL * DM;               // [4096*48]
  float* wq   = hT + (size_t)BATCH * HIDG; // qkvT [288][96]
  float* wo   = wq + 288 * 96;             // owT  [96][96]
  float* wf1  = wo + 96 * 96;              // f1T  [192][96]
  float* wf2  = wf1 + 192 * 96;            // f2T  [96][192]
  float* wpo  = wf2 + 96 * 192;            // postT [96][64]
  float* w1t1 = wpo + 96 * 64;             // [64][5][96]
  float* w2t1 = w1t1 + 64 * 5 * 96;        // [64][5][64]
  float* w1t2 = w2t1 + 64 * 5 * 64;
  float* w2t2 = w1t2 + 64 * 5 * 64;
  float* w1t3 = w2t2 + 64 * 5 * 64;
  float* w2t3 = w1t3 + 64 * 5 * 64;

  // --- weight prep ---
  auto TR = [&](const float* s, float* d, int K, int N) {
    int n = K * N;
    k_tr2d<<<(n + 255) / 256, 256, 0, stream>>>(s, d, K, N);
  };
  TR(F(LA_QKVW), wq, 96, 288);
  TR(F(LA_OW), wo, 96, 96);
  TR(F(LA_F1W), wf1, 96, 192);
  TR(F(LA_F2W), wf2, 192, 96);
  TR(F(IN_POSTW), wpo, 64, 96);
  k_trconv<<<(64 * 96 * 5 + 255) / 256, 256, 0, stream>>>(F(T1_W1), w1t1, 64, 96);
  k_trconv<<<(64 * 64 * 5 + 255) / 256, 256, 0, stream>>>(F(T1_W2), w2t1, 64, 64);
  k_trconv<<<(64 * 64 * 5 + 255) / 256, 256, 0, stream>>>(F(T2_W1), w1t2, 64, 64);
  k_trconv<<<(64 * 64 * 5 + 255) / 256, 256, 0, stream>>>(F(T2_W2), w2t2, 64, 64);
  k_trconv<<<(64 * 64 * 5 + 255) / 256, 256, 0, stream>>>(F(T3_W1), w1t3, 64, 64);
  k_trconv<<<(64 * 64 * 5 + 255) / 256, 256, 0, stream>>>(F(T3_W2), w2t3, 64, 64);

  // --- pipeline ---
  k_inproj<<<(int)(BL / 256), 256, 0, stream>>>(F(IN_X), F(IN_INW), F(IN_INB), A, (int)BL);

  constexpr int PAD = 16;
  k_tcn<96, 1, 2, true><<<BATCH, 128,
      ((PAD + SEQL) * 96 + (PAD + 80) * CT) * sizeof(float), stream>>>(
      A, Bb, F(T1_NG), F(T1_NB), w1t1, F(T1_B1), w2t1, F(T1_B2), F(T1_WRES), F(T1_BRES));
  k_tcn<64, 2, 4, false><<<BATCH, 128,
      ((PAD + SEQL) * 64 + (PAD + 80) * CT) * sizeof(float), stream>>>(
      Bb, C, F(T2_NG), F(T2_NB), w1t2, F(T2_B1), w2t2, F(T2_B2), nullptr, nullptr);
  k_tcn<64, 4, 4, false><<<BATCH, 128,
      ((PAD + SEQL) * 64 + (PAD + 80) * CT) * sizeof(float), stream>>>(
      C, Bb, F(T3_NG), F(T3_NB), w1t3, F(T3_B1), w2t3, F(T3_B2), nullptr, nullptr);

  k_post<<<(int)(BL / 16 / 8), 256, 8 * 16 * CT * sizeof(float), stream>>>(
      Bb, wpo, F(IN_POSTB), C);

  k_attn<<<BATCH, 32, 0, stream>>>(
      C, A, F(LA_LN1G), F(LA_LN1B), wq, F(LA_QKVB), wo, F(LA_OB),
      F(LA_LN2G), F(LA_LN2B), wf1, F(LA_F1B), wf2, F(LA_F2B),
      F(LA_PLNG), F(LA_PLNB));

  k_gru<<<BATCH / 16, 32, 0, stream>>>(A, F(GRU_WIH), F(GRU_WHH), F(GRU_BIH),
                                       F(GRU_BHH), hT);

  k_head<<<BATCH / 128, 128, 0, stream>>>(hT, F(H1W), F(H1B), F(H2W), F(H2B),
                                          (float*)d_out);
}